// GraphSAGE_72292889526467
// MI455X (gfx1250) — compile-verified
//
#include <hip/hip_runtime.h>

typedef __attribute__((ext_vector_type(2))) float    v2f;
typedef __attribute__((ext_vector_type(8))) float    v8f;
typedef __attribute__((ext_vector_type(4))) unsigned v4u;
typedef __attribute__((ext_vector_type(8))) int      v8i;
typedef __attribute__((ext_vector_type(4))) int      v4i;

#define SLI   8
#define FEAT  64
#define NODES 20000
#define NTOT  (SLI * NODES)
#define NEG_SLOPE 0.01f

__global__ void zero_f32(float* __restrict__ p, int n) {
  int i = blockIdx.x * blockDim.x + threadIdx.x;
  if (i < n) p[i] = 0.0f;
}

__global__ void degree_kernel(const int* __restrict__ src,
                              const int* __restrict__ dst,
                              float* __restrict__ deg_out,
                              float* __restrict__ deg_in, int E) {
  int e = blockIdx.x * blockDim.x + threadIdx.x;
  if (e < E) {
    atomicAdd(&deg_out[src[e]], 1.0f);
    atomicAdd(&deg_in[dst[e]], 1.0f);
  }
}

// in-place: deg -> (deg>0 ? deg^-0.5 : 1.0)
__global__ void norm_kernel(float* __restrict__ deg, int n) {
  int i = blockIdx.x * blockDim.x + threadIdx.x;
  if (i < n) {
    float v = deg[i];
    deg[i] = (v > 0.0f) ? rsqrtf(v) : 1.0f;
  }
}

// agg[dst] += x[src] * norm_src[src]; agg lives in d_out with [sli][feat][node] layout.
__global__ void scatter_kernel(const float* __restrict__ x,     // [sli][feat][node]
                               const int* __restrict__ src,
                               const int* __restrict__ dst,
                               const float* __restrict__ norm_src,
                               float* __restrict__ agg, int E) {
  unsigned t = blockIdx.x * blockDim.x + threadIdx.x;
  unsigned e = t >> 6;
  unsigned f = t & 63u;
  if (e < (unsigned)E) {
    int s = src[e];
    int d = dst[e];
    int ss = s / NODES, sn = s - ss * NODES;
    int ds = d / NODES, dn = d - ds * NODES;
    float val = x[(ss * FEAT + (int)f) * NODES + sn] * norm_src[s];
    atomicAdd(&agg[(ds * FEAT + (int)f) * NODES + dn], val);
  }
}

// One wave32 per 16-node tile: D = leaky_relu((norm_dst ⊙ agg) @ W + b)
// io holds agg on entry, the final transposed result on exit (tile regions are
// self-contained: each wave reads exactly the 16x64 region it later writes,
// ordered by the register RAW chain through the WMMA accumulators).
__global__ void __launch_bounds__(256)
gconv_wmma_kernel(float* __restrict__ io,          // [sli][64][node]
                  const float* __restrict__ W,     // [64][64] (k, col) row-major
                  const float* __restrict__ bias,  // [64]
                  const float* __restrict__ norm_dst) {
  __shared__ float sW[FEAT * FEAT];                // 16 KB of the 320 KB WGP LDS

  // ---- stage W into LDS once per workgroup via the Tensor Data Mover ----
#if __has_builtin(__builtin_amdgcn_tensor_load_to_lds) && \
    __has_builtin(__builtin_amdgcn_s_wait_tensorcnt)
  if (threadIdx.x < 32) {                          // wave 0 drives the TDM
    unsigned lds_addr =
        (unsigned)(unsigned long long)(__attribute__((address_space(3))) float*)sW;
    unsigned long long ga = (unsigned long long)W;
    // D# group 0: count=1 (valid), lds_addr, 57-bit global addr, type=2
    v4u g0 = { 1u,
               lds_addr,
               (unsigned)(ga & 0xFFFFFFFFull),
               (unsigned)((ga >> 32) & 0x01FFFFFFull) | (2u << 30) };
    // D# group 1: 1-D tensor of 4096 x 4-byte elements, tile = whole tensor
    v8i g1 = { 0x20000,               // data_size=2 (4B); no multicast/barrier/pad
               (int)(4096u << 16),    // tensor_dim0[15:0] in bits 63:48
               0x00010000,            // tensor_dim0[31:16]=0, tensor_dim1=1
               (int)(4096u << 16),    // tensor_dim1[31:16]=0, tile_dim0=4096
               1,                     // tile_dim1=1, tile_dim2=0
               4096,                  // tensor_dim0_stride[31:0]
               0, 0 };                // stride hi, tensor_dim1_stride
    v4i gz4 = {0, 0, 0, 0};           // groups 2/3 unused (<=2D tensor)
    v8i gz8 = {0, 0, 0, 0, 0, 0, 0, 0};
    // 6-arg toolchain form: (g0, g1, g2, g3, g4, cpol)
    __builtin_amdgcn_tensor_load_to_lds(g0, g1, gz4, gz4, gz8, 0);
  }
  __builtin_amdgcn_s_wait_tensorcnt(0);            // TENSORcnt==0 (NOP for waves 1..7)
  __syncthreads();
#else
  for (int i = threadIdx.x; i < FEAT * FEAT; i += 256) sW[i] = W[i];
  __syncthreads();
#endif

  const int lane  = threadIdx.x & 31;
  const int wave  = (blockIdx.x * blockDim.x + threadIdx.x) >> 5;
  const int base  = wave * 16;              // flattened node base (16 | 20000)
  const int sli   = base / NODES;
  const int node0 = base - sli * NODES;

  const int rowA = lane & 15;               // A-matrix M for this lane
  const int k0   = (lane >> 4) << 1;        // K sub-offset: lanes 0-15 -> 0, 16-31 -> 2
  const int hi   = lane >> 4;

  // ---- load full 16x64 A tile into registers, fold in norm_dst row scale ----
  const float ns = norm_dst[base + rowA];
  const float* aggS = io + (size_t)sli * FEAT * NODES + node0 + rowA;
  v2f a[16];
#pragma unroll
  for (int kk = 0; kk < 16; ++kk) {
    int f = (kk << 2) + k0;
    float x0 = aggS[(size_t)f * NODES];
    float x1 = aggS[(size_t)(f + 1) * NODES];
    a[kk].x = x0 * ns;
    a[kk].y = x1 * ns;
  }

  const int colL = lane & 15;
#pragma unroll
  for (int n = 0; n < 4; ++n) {
    const int col = (n << 4) + colL;
    v8f c = {0.f, 0.f, 0.f, 0.f, 0.f, 0.f, 0.f, 0.f};
#pragma unroll
    for (int kk = 0; kk < 16; ++kk) {
      const int krow = (kk << 2) + k0;
      v2f bmat;
      bmat.x = sW[krow * 64 + col];         // ds_load from LDS-resident W
      bmat.y = sW[krow * 64 + 64 + col];
      // D = A(16x4,f32) x B(4x16,f32) + C
      c = __builtin_amdgcn_wmma_f32_16x16x4_f32(false, a[kk], false, bmat,
                                                (short)0, c, false, false);
    }
    const float bb = bias[col];
    float* outc = io + ((size_t)sli * FEAT + col) * NODES + node0 + (hi << 3);
    float r0 = c[0] + bb, r1 = c[1] + bb, r2 = c[2] + bb, r3 = c[3] + bb;
    float r4 = c[4] + bb, r5 = c[5] + bb, r6 = c[6] + bb, r7 = c[7] + bb;
    r0 = r0 > 0.f ? r0 : NEG_SLOPE * r0;  r1 = r1 > 0.f ? r1 : NEG_SLOPE * r1;
    r2 = r2 > 0.f ? r2 : NEG_SLOPE * r2;  r3 = r3 > 0.f ? r3 : NEG_SLOPE * r3;
    r4 = r4 > 0.f ? r4 : NEG_SLOPE * r4;  r5 = r5 > 0.f ? r5 : NEG_SLOPE * r5;
    r6 = r6 > 0.f ? r6 : NEG_SLOPE * r6;  r7 = r7 > 0.f ? r7 : NEG_SLOPE * r7;
    // 16B-aligned: node0 % 16 == 0, 20000*4B % 16 == 0, hi*8*4B % 16 == 0
    *(float4*)(outc)     = make_float4(r0, r1, r2, r3);   // global_store_b128
    *(float4*)(outc + 4) = make_float4(r4, r5, r6, r7);
  }
}

extern "C" void kernel_launch(void* const* d_in, const int* in_sizes, int n_in,
                              void* d_out, int out_size, void* d_ws, size_t ws_size,
                              hipStream_t stream) {
  const float* x    = (const float*)d_in[0];   // [8][64][20000]
  const float* W    = (const float*)d_in[1];   // [64][64]
  const float* bias = (const float*)d_in[2];   // [64]
  const int*   src  = (const int*)d_in[3];
  const int*   dst  = (const int*)d_in[4];
  float* out = (float*)d_out;

  const int E = in_sizes[3];
  const int N = NTOT;

  float* deg_out = (float*)d_ws;               // N floats; deg_in = deg_out + N

  // zero agg (= d_out) and degree counters
  zero_f32<<<(out_size + 255) / 256, 256, 0, stream>>>(out, out_size);
  zero_f32<<<(2 * N + 255) / 256, 256, 0, stream>>>(deg_out, 2 * N);

  degree_kernel<<<(E + 255) / 256, 256, 0, stream>>>(src, dst, deg_out, deg_out + N, E);
  norm_kernel<<<(2 * N + 255) / 256, 256, 0, stream>>>(deg_out, 2 * N);

  long long scat_threads = (long long)E * 64;
  scatter_kernel<<<(unsigned)((scat_threads + 255) / 256), 256, 0, stream>>>(
      x, src, dst, /*norm_src=*/deg_out, out, E);

  const int tiles = N / 16;                    // 10000 wave-tiles
  gconv_wmma_kernel<<<(tiles * 32) / 256, 256, 0, stream>>>(
      out, W, bias, /*norm_dst=*/deg_out + N);
}